// CenterModel_15092515078257
// MI455X (gfx1250) — compile-verified
//
#include <hip/hip_runtime.h>
#include <hip/hip_bf16.h>
#include <math.h>

// ---------------------------------------------------------------------------
// CenterModel metric on MI455X (gfx1250, wave32).
//
// Phase 1 (memory-bound, ~11us floor @ 23.3 TB/s for the 256MB tid stream):
//   grid-stride over points; lane w (0..127) owns wavelength w so each row's
//   128 floats of `tid` are one fully coalesced 512B transaction. tid is
//   single-use and ~exceeds the 192MB L2, so load it non-temporally.
//   Per-block partial sums (C[w], S[w]) go to d_ws -> deterministic (no
//   float atomics).
//
// Phase 2 (finalize): vals[128x50] = [C S](128x2) x [cos(o);-sin(o)](2x50)
//   is a real matmul -> V_WMMA_F32_16X16X4_F32 (K padded 2->4 with zeros).
//   8 waves x 4 offset-tiles of 16 columns, then per-row max via 16-lane
//   shfl_xor butterfly (invalid padded columns masked to -inf), sum, negate.
// ---------------------------------------------------------------------------

#define WN 128          // number of wavelengths (lanes in w-dimension)
#define ROWS_PER_BLK 4  // row slices per block (512 threads = 128w x 4rows)
#define NBLK 512        // partial-sum blocks (512*128*2*4B = 512KB in d_ws)

#define TWO_PI 6.28318530717958647692f
#define NEG_HUGE (-3.402823466e38f)

typedef __attribute__((ext_vector_type(2))) float v2f;
typedef __attribute__((ext_vector_type(8))) float v8f;

// ---------------------------------------------------------------- phase 1 --
__global__ __launch_bounds__(512) void cm_partials(
    const float* __restrict__ xy, const float* __restrict__ tidm,
    const float* __restrict__ center, const float* __restrict__ wavelength,
    float* __restrict__ pC, float* __restrict__ pS, int npts)
{
    const int t    = threadIdx.x;
    const int w    = t & (WN - 1);   // lane's wavelength column
    const int ysub = t >> 7;         // row slice 0..3 (uniform per wave)

    const float cx = center[0];
    const float cy = center[1];
    const float k  = TWO_PI / wavelength[w];   // 2*pi / lambda_w

    float accC = 0.0f, accS = 0.0f;

    for (int n = blockIdx.x * ROWS_PER_BLK + ysub; n < npts;
         n += NBLK * ROWS_PER_BLK) {
        // xy row is uniform across the wave's 32 lanes -> single cacheline.
        float dx = xy[2 * n + 0] - cx;
        float dy = xy[2 * n + 1] - cy;
        float dist = sqrtf(fmaf(dx, dx, dy * dy));
        float ph = k * dist;
        float s, c;
        __sincosf(ph, &s, &c);
        // Streamed once, bigger than L2 -> non-temporal.
        float tv = __builtin_nontemporal_load(&tidm[(size_t)n * WN + w]);
        accC = fmaf(c, tv, accC);
        accS = fmaf(s, tv, accS);
    }

    __shared__ float sC[ROWS_PER_BLK][WN];
    __shared__ float sS[ROWS_PER_BLK][WN];
    sC[ysub][w] = accC;
    sS[ysub][w] = accS;
    __syncthreads();

    if (t < WN) {
        float c = sC[0][t] + sC[1][t] + sC[2][t] + sC[3][t];
        float s = sS[0][t] + sS[1][t] + sS[2][t] + sS[3][t];
        pC[(size_t)blockIdx.x * WN + t] = c;   // coalesced across t
        pS[(size_t)blockIdx.x * WN + t] = s;
    }
}

// ---------------------------------------------------------------- phase 2 --
// One block, 256 threads = 8 waves. Wave wv owns w-rows [16*wv, 16*wv+16).
__global__ __launch_bounds__(256) void cm_finalize(
    const float* __restrict__ pC, const float* __restrict__ pS,
    float* __restrict__ out, float invN)
{
    __shared__ float lC[WN];
    __shared__ float lS[WN];
    __shared__ float wsums[8];

    const int t = threadIdx.x;

    // Cross-block partial reduction (coalesced: fixed b, consecutive t).
    if (t < WN) {
        float c = 0.0f, s = 0.0f;
        for (int b = 0; b < NBLK; ++b) {
            c += pC[(size_t)b * WN + t];
            s += pS[(size_t)b * WN + t];
        }
        lC[t] = c * invN;   // mean over points
        lS[t] = s * invN;
    }
    __syncthreads();

    const int wv   = t >> 5;       // wave id = w-tile id (0..7)
    const int lane = t & 31;
    const int half = lane & 15;    // output column index within tile
    const bool hi  = (lane >= 16); // upper half carries K=2,3 (zero pad)

    // A fragment, 16x4 f32 (M=w rows, K = {C, S, 0, 0}):
    //   lanes 0-15:  VGPR0 = A[m][0] = C[w], VGPR1 = A[m][1] = S[w]
    //   lanes 16-31: VGPR0 = A[m][2] = 0,    VGPR1 = A[m][3] = 0
    v2f a;
    a.x = hi ? 0.0f : lC[wv * 16 + half];
    a.y = hi ? 0.0f : lS[wv * 16 + half];

    float mmax[8];
#pragma unroll
    for (int r = 0; r < 8; ++r) mmax[r] = NEG_HUGE;

#pragma unroll
    for (int nt = 0; nt < 4; ++nt) {     // 4 tiles of 16 offsets cover 50
        int j  = nt * 16 + half;         // global offset index (column N)
        int jc = (j < 50) ? j : 49;      // clamp for padded columns
        float o = TWO_PI * (float)jc * (1.0f / 49.0f);  // linspace(0,2pi,50)

        // B fragment, 4x16 f32 (K rows striped across lane halves):
        //   lanes 0-15:  VGPR0 = B[0][n] = cos(o_n), VGPR1 = B[1][n] = -sin(o_n)
        //   lanes 16-31: K=2,3 rows = 0
        v2f bf;
        bf.x = hi ? 0.0f : __cosf(o);
        bf.y = hi ? 0.0f : -__sinf(o);

        v8f cz = {};
        // D(16x16) = A(16x4) x B(4x16) + 0 : vals[w_tile, offset_tile]
        v8f d = __builtin_amdgcn_wmma_f32_16x16x4_f32(
            /*neg_a=*/false, a, /*neg_b=*/false, bf,
            /*c_mod=*/(short)0, cz, /*reuse_a=*/false, /*reuse_b=*/false);

        const bool valid = (j < 50);
#pragma unroll
        for (int r = 0; r < 8; ++r) {
            // lanes 0-15 hold row M=r (w = 16*wv + r), lanes 16-31 row M=8+r.
            float v = valid ? d[r] : NEG_HUGE;
            // max over the 16 columns of this tile (stays within each half)
            v = fmaxf(v, __shfl_xor(v, 1, 32));
            v = fmaxf(v, __shfl_xor(v, 2, 32));
            v = fmaxf(v, __shfl_xor(v, 4, 32));
            v = fmaxf(v, __shfl_xor(v, 8, 32));
            mmax[r] = fmaxf(mmax[r], v);
        }
    }

    // Sum the per-row maxima: lane 0 holds rows M=0..7, lane 16 rows M=8..15.
    float wsum = 0.0f;
#pragma unroll
    for (int r = 0; r < 8; ++r) wsum += mmax[r];

    float s0  = __shfl(wsum, 0, 32);
    float s16 = __shfl(wsum, 16, 32);
    if (lane == 0) wsums[wv] = s0 + s16;
    __syncthreads();

    if (t == 0) {
        float total = 0.0f;
        for (int i = 0; i < 8; ++i) total += wsums[i];
        out[0] = -total;   // reference returns -metric
    }
}

// ------------------------------------------------------------------ launch --
extern "C" void kernel_launch(void* const* d_in, const int* in_sizes, int n_in,
                              void* d_out, int out_size, void* d_ws, size_t ws_size,
                              hipStream_t stream) {
    const float* xy     = (const float*)d_in[0];   // [N,2]
    const float* tidm   = (const float*)d_in[1];   // [N,128]
    const float* center = (const float*)d_in[2];   // [2]
    const float* wl     = (const float*)d_in[3];   // [128]
    float* out          = (float*)d_out;

    const int npts = in_sizes[0] / 2;

    float* pC = (float*)d_ws;                 // [NBLK, 128]
    float* pS = pC + (size_t)NBLK * WN;       // [NBLK, 128]

    cm_partials<<<NBLK, 512, 0, stream>>>(xy, tidm, center, wl, pC, pS, npts);
    cm_finalize<<<1, 256, 0, stream>>>(pC, pS, out, 1.0f / (float)npts);
}